// NexToU_Encoder_42941083025877
// MI455X (gfx1250) — compile-verified
//
#include <hip/hip_runtime.h>

// ---------------- problem constants ----------------
#define B_   2
#define C_   96
#define D_   16
#define H_   64
#define W_   64
#define S_   (D_*H_*W_)          // 65536 voxels per batch
#define NWIN 512                 // B * (16/4)*(64/8)*(64/8)
#define NNODE 256                // 4*8*8
#define KNN  9
#define NEG_SLOPE 0.2f
#define EPS  1e-5f

typedef float v2f __attribute__((ext_vector_type(2)));
typedef float v8f __attribute__((ext_vector_type(8)));

// =====================================================================
// fp32 WMMA GEMM:  Y[M,P] = Wm[M,K] @ X[K,P] + bias
// X/Y row stride == P; batched over grid.z (element strides bsx/bsy).
// One wave computes a 16x64 strip: 4 accumulators share one A fragment.
// All 9 loads of a K-step are issued before the 4 WMMAs so the backend
// can clause them and overlap latency with the WMMA pipe.
// P must be a multiple of 64 (65536 and 256 here both are).
// =====================================================================
__global__ __launch_bounds__(256) void gemm_wmma(
    const float* __restrict__ Wm, const float* __restrict__ X,
    const float* __restrict__ bias, float* __restrict__ Y,
    int M, int K, int P, long bsx, long bsy)
{
    int wid  = (blockIdx.x * blockDim.x + threadIdx.x) >> 5;
    int lane = threadIdx.x & 31;
    int Mt = M >> 4, Pt4 = P >> 6;
    if (wid >= Mt * Pt4) return;                // wave-uniform exit (EXEC all-1s kept)
    int m0 = (wid % Mt) << 4;
    int n0 = (wid / Mt) << 6;
    const float* Xb = X + (long)blockIdx.z * bsx;
    float*       Yb = Y + (long)blockIdx.z * bsy;
    int half = lane >> 4, l16 = lane & 15;

    // A (16x4): lane l16 -> row M=m0+l16 ; vgpr v + 2*half -> K offset
    const float* wrow = Wm + (long)(m0 + l16) * K + 2 * half;
    // B (4x16): lane l16 -> col ; vgpr v + 2*half -> K offset
    const float* xcol = Xb + (long)(2 * half) * P + n0 + l16;

    v8f acc[4];
    #pragma unroll
    for (int t = 0; t < 4; ++t) acc[t] = (v8f){0.f,0.f,0.f,0.f,0.f,0.f,0.f,0.f};

    for (int k0 = 0; k0 < K; k0 += 4) {
        const float* xp = xcol + (long)k0 * P;
        // ---- load phase: distinct registers, clause-able ----
        v2f a, bb[4];
        a.x = wrow[k0];
        a.y = wrow[k0 + 1];
        #pragma unroll
        for (int t = 0; t < 4; ++t) {
            bb[t].x = xp[t * 16];
            bb[t].y = xp[P + t * 16];
        }
        // ---- compute phase: 4 independent WMMAs back-to-back ----
        #pragma unroll
        for (int t = 0; t < 4; ++t) {
            acc[t] = __builtin_amdgcn_wmma_f32_16x16x4_f32(
                false, a, false, bb[t], (short)0, acc[t], false, false);
        }
    }
    // D (16x16): vgpr v, lanes 0-15 -> M=m0+v ; lanes 16-31 -> M=m0+v+8
    #pragma unroll
    for (int v = 0; v < 8; ++v) {
        int m = m0 + v + 8 * half;
        float bi = bias[m];
        float* yr = Yb + (long)m * P + n0 + l16;
        #pragma unroll
        for (int t = 0; t < 4; ++t) yr[t * 16] = acc[t][v] + bi;
    }
}

// =====================================================================
// GroupNorm statistics over contiguous group regions (atomic partials)
// grid.x = total groups, grid.y = slices per group
// =====================================================================
__global__ __launch_bounds__(256) void gn_stats(
    const float* __restrict__ X, float* __restrict__ Stats, long groupElems)
{
    long base = (long)blockIdx.x * groupElems;
    long per  = (groupElems + gridDim.y - 1) / gridDim.y;
    long s = (long)blockIdx.y * per;
    long e = s + per; if (e > groupElems) e = groupElems;
    float sum = 0.f, sq = 0.f;
    for (long i = s + threadIdx.x; i < e; i += 256) {
        float v = X[base + i];
        sum += v; sq += v * v;
    }
    __shared__ float r0[256], r1[256];
    r0[threadIdx.x] = sum; r1[threadIdx.x] = sq;
    __syncthreads();
    for (int off = 128; off > 0; off >>= 1) {
        if (threadIdx.x < off) { r0[threadIdx.x] += r0[threadIdx.x+off]; r1[threadIdx.x] += r1[threadIdx.x+off]; }
        __syncthreads();
    }
    if (threadIdx.x == 0) {
        atomicAdd(&Stats[2*blockIdx.x],   r0[0]);
        atomicAdd(&Stats[2*blockIdx.x+1], r1[0]);
    }
}

// GroupNorm apply: out = (x-mu)*rsqrt(var+eps)*g + b ; optional lrelu/residual
__global__ __launch_bounds__(256) void gn_apply(
    const float* __restrict__ Xin, float* __restrict__ Out,
    const float* __restrict__ stats, const float* __restrict__ gamma,
    const float* __restrict__ beta, const float* __restrict__ resid,
    int doLrelu, int Cc, int logL, int gsize, unsigned total, float invDenom)
{
    unsigned i = blockIdx.x * 256u + threadIdx.x;
    if (i >= total) return;
    unsigned rowIdx = i >> logL;
    int c  = rowIdx % Cc;
    int g  = (rowIdx / Cc) * (Cc / gsize) + c / gsize;
    float mu  = stats[2*g]   * invDenom;
    float var = stats[2*g+1] * invDenom - mu * mu;
    float v = (Xin[i] - mu) * rsqrtf(var + EPS) * gamma[c] + beta[c];
    if (doLrelu) v = (v >= 0.f) ? v : NEG_SLOPE * v;
    if (resid)   v += resid[i];
    Out[i] = v;
}

// =====================================================================
// Window partition: (B,C,D,H,W) -> low 96 channels of xcat (NWIN,192,256)
// =====================================================================
__global__ __launch_bounds__(256) void win_partition(
    const float* __restrict__ Xin, float* __restrict__ Xcat)
{
    unsigned i = blockIdx.x * 256u + threadIdx.x;
    if (i >= (unsigned)NWIN * C_ * NNODE) return;
    unsigned wi = i / (C_ * NNODE);
    unsigned rem = i % (C_ * NNODE);
    int c = rem >> 8, n = rem & 255;
    int b = wi >> 8, wd = (wi >> 6) & 3, wh = (wi >> 3) & 7, ww = wi & 7;
    int pd = n >> 6, ph = (n >> 3) & 7, pw = n & 7;
    int d = wd*4 + pd, h = wh*8 + ph, w = ww*8 + pw;
    long src = (((long)(b * C_ + c) * D_ + d) * H_ + h) * W_ + w;
    Xcat[(long)wi * (192*NNODE) + c * NNODE + n] = Xin[src];
}

// =====================================================================
// KNN per window: top-9 by neg squared distance (= 2*inner - sq_n - sq_m)
// One block / window, thread n owns row n (96 floats in VGPRs),
// 32-row tiles of opposing rows staged in LDS.
// =====================================================================
__global__ __launch_bounds__(256) void knn_kernel(
    const float* __restrict__ Xcat, int* __restrict__ Idx)
{
    int wi = blockIdx.x;
    int n  = threadIdx.x;
    const float* fw = Xcat + (long)wi * (192*NNODE);   // first 96 rows = xw

    float myf[96];
    #pragma unroll
    for (int c = 0; c < 96; ++c) myf[c] = fw[c * 256 + n];
    float sq = 0.f;
    #pragma unroll
    for (int c = 0; c < 96; ++c) sq += myf[c] * myf[c];

    __shared__ float ssq[256];
    __shared__ float sf[32][97];
    ssq[n] = sq;

    float tv[9]; int ti[9];
    #pragma unroll
    for (int k = 0; k < 9; ++k) { tv[k] = -3.4e38f; ti[k] = 0; }
    __syncthreads();

    for (int mb = 0; mb < 8; ++mb) {
        for (int j = n; j < 32*96; j += 256) {    // coalesced tile load
            int r = j & 31, c = j >> 5;
            sf[r][c] = fw[c * 256 + mb*32 + r];
        }
        __syncthreads();
        for (int r = 0; r < 32; ++r) {
            int m = mb*32 + r;
            float dot = 0.f;
            #pragma unroll
            for (int c = 0; c < 96; ++c) dot += myf[c] * sf[r][c];
            float nd = 2.f * dot - sq - ssq[m];
            if (nd > tv[8]) {                     // stable insert (ties keep lower m)
                tv[8] = nd; ti[8] = m;
                #pragma unroll
                for (int j = 8; j > 0; --j) {
                    if (tv[j] > tv[j-1]) {
                        float t = tv[j]; tv[j] = tv[j-1]; tv[j-1] = t;
                        int   q = ti[j]; ti[j] = ti[j-1]; ti[j-1] = q;
                    }
                }
            }
        }
        __syncthreads();
    }
    #pragma unroll
    for (int k = 0; k < 9; ++k) Idx[(long)wi * (NNODE*KNN) + n*KNN + k] = ti[k];
}

// Neighbor-count histogram per window (gather multiplicity for weighted GN stats)
__global__ __launch_bounds__(256) void hist_kernel(
    const int* __restrict__ Idx, int* __restrict__ Cnt)
{
    unsigned i = blockIdx.x * 256u + threadIdx.x;
    if (i >= (unsigned)NWIN * NNODE * KNN) return;
    unsigned wi = i / (NNODE * KNN);
    atomicAdd(&Cnt[wi * NNODE + Idx[i]], 1);
}

// Count-weighted GN stats of z == stats of gathered x_j (gather commutes w/ conv)
__global__ __launch_bounds__(256) void wstats_kernel(
    const float* __restrict__ Z, const int* __restrict__ Cnt, float* __restrict__ Stats)
{
    int g  = blockIdx.x;              // NWIN*16 blocks
    int wi = g >> 4, grp = g & 15;
    const float* zb = Z + (long)wi * (C_*NNODE) + grp * 6 * NNODE;
    const int*   cb = Cnt + wi * NNODE;
    float sum = 0.f, sq = 0.f;
    for (int j = threadIdx.x; j < 6*NNODE; j += 256) {
        int c = j >> 8, m = j & 255;
        float w = (float)cb[m];
        float v = zb[c * NNODE + m];
        sum += w * v; sq += w * v * v;
    }
    __shared__ float r0[256], r1[256];
    r0[threadIdx.x] = sum; r1[threadIdx.x] = sq;
    __syncthreads();
    for (int off = 128; off > 0; off >>= 1) {
        if (threadIdx.x < off) { r0[threadIdx.x] += r0[threadIdx.x+off]; r1[threadIdx.x] += r1[threadIdx.x+off]; }
        __syncthreads();
    }
    if (threadIdx.x == 0) { Stats[2*g] = r0[0]; Stats[2*g+1] = r1[0]; }
}

// gather 9 neighbors of z, GN+lrelu each, max -> high 96 channels of xcat
__global__ __launch_bounds__(256) void aggregate_kernel(
    const float* __restrict__ Z, const int* __restrict__ Idx,
    const float* __restrict__ Stats, const float* __restrict__ gamma,
    const float* __restrict__ beta, float* __restrict__ Xcat)
{
    unsigned i = blockIdx.x * 256u + threadIdx.x;
    if (i >= (unsigned)NWIN * C_ * NNODE) return;
    unsigned wi = i / (C_*NNODE);
    unsigned rem = i % (C_*NNODE);
    int c = rem >> 8, n = rem & 255;
    int g = wi * 16 + c / 6;
    const float invD = 1.f / (6.f * NNODE * KNN);        // 1/13824
    float mu  = Stats[2*g] * invD;
    float var = Stats[2*g+1] * invD - mu * mu;
    float scl = rsqrtf(var + EPS) * gamma[c];
    float bt  = beta[c];
    const float* zb = Z + (long)wi * (C_*NNODE) + c * NNODE;
    const int*   ib = Idx + (long)wi * (NNODE*KNN) + n * KNN;
    float mx = -3.4e38f;
    #pragma unroll
    for (int k = 0; k < 9; ++k) {
        float v = (zb[ib[k]] - mu) * scl + bt;
        v = (v >= 0.f) ? v : NEG_SLOPE * v;
        mx = fmaxf(mx, v);
    }
    Xcat[(long)wi * (192*NNODE) + (C_ + c) * NNODE + n] = mx;
}

// GN+lrelu on z2 fused with window_reverse -> (B,192,D,H,W)
__global__ __launch_bounds__(256) void gn_apply_reverse(
    const float* __restrict__ Z2, const float* __restrict__ Stats,
    const float* __restrict__ gamma, const float* __restrict__ beta,
    float* __restrict__ Out)
{
    unsigned i = blockIdx.x * 256u + threadIdx.x;
    if (i >= (unsigned)NWIN * 192 * NNODE) return;
    unsigned wi = i / (192*NNODE);
    unsigned rem = i % (192*NNODE);
    int c = rem >> 8, n = rem & 255;
    int g = wi * 32 + c / 6;
    const float invD = 1.f / (6.f * NNODE);              // 1/1536
    float mu  = Stats[2*g] * invD;
    float var = Stats[2*g+1] * invD - mu * mu;
    float v = (Z2[i] - mu) * rsqrtf(var + EPS) * gamma[c] + beta[c];
    v = (v >= 0.f) ? v : NEG_SLOPE * v;
    int b = wi >> 8, wd = (wi >> 6) & 3, wh = (wi >> 3) & 7, ww = wi & 7;
    int pd = n >> 6, ph = (n >> 3) & 7, pw = n & 7;
    int d = wd*4 + pd, h = wh*8 + ph, w = ww*8 + pw;
    long off = (((long)(b * 192 + c) * D_ + d) * H_ + h) * W_ + w;
    Out[off] = v;
}

// =====================================================================
// Host-side orchestration
// =====================================================================
static inline dim3 grid1(long total) { return dim3((unsigned)((total + 255) / 256)); }
static inline dim3 gemm_grid(int M, int P, int z) {
    long waves = (long)(M >> 4) * (P >> 6);   // one wave per 16x64 strip
    return dim3((unsigned)((waves * 32 + 255) / 256), 1, (unsigned)z);
}

extern "C" void kernel_launch(void* const* d_in, const int* in_sizes, int n_in,
                              void* d_out, int out_size, void* d_ws, size_t ws_size,
                              hipStream_t stream) {
    const float* x      = (const float*)d_in[0];
    const float* fc1_w  = (const float*)d_in[1];  const float* fc1_b  = (const float*)d_in[2];
    const float* fc1_g  = (const float*)d_in[3];  const float* fc1_bt = (const float*)d_in[4];
    const float* nn1_w  = (const float*)d_in[5];  const float* nn1_b  = (const float*)d_in[6];
    const float* nn1_g  = (const float*)d_in[7];  const float* nn1_bt = (const float*)d_in[8];
    const float* nn2_w  = (const float*)d_in[9];  const float* nn2_b  = (const float*)d_in[10];
    const float* nn2_g  = (const float*)d_in[11]; const float* nn2_bt = (const float*)d_in[12];
    const float* fc2_w  = (const float*)d_in[13]; const float* fc2_b  = (const float*)d_in[14];
    const float* fc2_g  = (const float*)d_in[15]; const float* fc2_bt = (const float*)d_in[16];
    const float* ffn1_w = (const float*)d_in[17]; const float* ffn1_b = (const float*)d_in[18];
    const float* ffn1_g = (const float*)d_in[19]; const float* ffn1_bt= (const float*)d_in[20];
    const float* ffn2_w = (const float*)d_in[21]; const float* ffn2_b = (const float*)d_in[22];
    const float* ffn2_g = (const float*)d_in[23]; const float* ffn2_bt= (const float*)d_in[24];
    float* out = (float*)d_out;

    const long CS  = (long)C_ * S_;          // 6,291,456
    const long C2S = 2L * C_ * S_;           // 12,582,912

    // workspace layout (bytes)
    char* wsb = (char*)d_ws;
    float* R1    = (float*)(wsb + 0);                 //  50.3 MB : y1 / z / y2
    float* Xcat  = (float*)(wsb + 50331648);          // 100.7 MB : xcat / xg_full / h
    float* R4    = (float*)(wsb + 150994944);         // 100.7 MB : z2 / xres
    int* IdxB    = (int*)  (wsb + 251658240);         //   4.7 MB
    int* CntB    = (int*)  (wsb + 256376832);         //   0.5 MB
    float* StatA = (float*)(wsb + 256901120);         // 128 KB (atomic stats)
    float* StatW = (float*)(wsb + 257032192);         //  64 KB (weighted stats)

    // ---- 1) fc1 conv + GN ------------------------------------------------
    gemm_wmma<<<gemm_grid(C_, S_, B_), 256, 0, stream>>>(fc1_w, x, fc1_b, R1, C_, C_, S_, CS, CS);
    hipMemsetAsync(StatA, 0, 32*2*sizeof(float), stream);
    gn_stats<<<dim3(32, 32), 256, 0, stream>>>(R1, StatA, 6L*S_);
    gn_apply<<<grid1(B_*CS), 256, 0, stream>>>(R1, R1, StatA, fc1_g, fc1_bt, nullptr,
                                               0, C_, 16, 6, (unsigned)(B_*CS), 1.f/(6.f*S_));
    // ---- 2) window partition + KNN --------------------------------------
    win_partition<<<grid1((long)NWIN*C_*NNODE), 256, 0, stream>>>(R1, Xcat);
    knn_kernel<<<NWIN, 256, 0, stream>>>(Xcat, IdxB);
    // ---- 3) nn1 conv (z), weighted GN stats, gather/lrelu/max -----------
    gemm_wmma<<<gemm_grid(C_, NNODE, NWIN), 256, 0, stream>>>(nn1_w, Xcat, nn1_b, R1,
        C_, C_, NNODE, 192L*NNODE, (long)C_*NNODE);
    hipMemsetAsync(CntB, 0, NWIN*NNODE*sizeof(int), stream);
    hist_kernel<<<grid1((long)NWIN*NNODE*KNN), 256, 0, stream>>>(IdxB, CntB);
    wstats_kernel<<<NWIN*16, 256, 0, stream>>>(R1, CntB, StatW);
    aggregate_kernel<<<grid1((long)NWIN*C_*NNODE), 256, 0, stream>>>(R1, IdxB, StatW, nn1_g, nn1_bt, Xcat);
    // ---- 4) nn2 conv + GN + lrelu + window_reverse ----------------------
    gemm_wmma<<<gemm_grid(192, NNODE, NWIN), 256, 0, stream>>>(nn2_w, Xcat, nn2_b, R4,
        192, 192, NNODE, 192L*NNODE, 192L*NNODE);
    hipMemsetAsync(StatA, 0, NWIN*32*2*sizeof(float), stream);
    gn_stats<<<dim3(NWIN*32, 1), 256, 0, stream>>>(R4, StatA, 6L*NNODE);
    gn_apply_reverse<<<grid1((long)NWIN*192*NNODE), 256, 0, stream>>>(R4, StatA, nn2_g, nn2_bt, Xcat);
    // ---- 5) fc2 conv + GN + residual(x) -> xres -------------------------
    gemm_wmma<<<gemm_grid(C_, S_, B_), 256, 0, stream>>>(fc2_w, Xcat, fc2_b, R1,
        C_, 192, S_, C2S, CS);
    hipMemsetAsync(StatA, 0, 32*2*sizeof(float), stream);
    gn_stats<<<dim3(32, 32), 256, 0, stream>>>(R1, StatA, 6L*S_);
    float* Xres = R4;
    gn_apply<<<grid1(B_*CS), 256, 0, stream>>>(R1, Xres, StatA, fc2_g, fc2_bt, x,
                                               0, C_, 16, 6, (unsigned)(B_*CS), 1.f/(6.f*S_));
    // ---- 6) FFN, processed per batch (reuses 100MB Xcat region for h) ---
    for (int b = 0; b < B_; ++b) {
        const float* xr = Xres + (long)b * CS;
        float* h = Xcat;
        gemm_wmma<<<gemm_grid(4*C_, S_, 1), 256, 0, stream>>>(ffn1_w, xr, ffn1_b, h,
            4*C_, C_, S_, 0, 0);
        hipMemsetAsync(StatA, 0, 64*2*sizeof(float), stream);
        gn_stats<<<dim3(64, 32), 256, 0, stream>>>(h, StatA, 6L*S_);
        gn_apply<<<grid1(4L*C_*S_), 256, 0, stream>>>(h, h, StatA, ffn1_g, ffn1_bt, nullptr,
                                                      1, 4*C_, 16, 6, (unsigned)(4L*C_*S_), 1.f/(6.f*S_));
        float* yb = out + (long)b * CS;
        gemm_wmma<<<gemm_grid(C_, S_, 1), 256, 0, stream>>>(ffn2_w, h, ffn2_b, yb,
            C_, 4*C_, S_, 0, 0);
        hipMemsetAsync(StatA, 0, 16*2*sizeof(float), stream);
        gn_stats<<<dim3(16, 32), 256, 0, stream>>>(yb, StatA, 6L*S_);
        gn_apply<<<grid1(CS), 256, 0, stream>>>(yb, yb, StatA, ffn2_g, ffn2_bt, xr,
                                                0, C_, 16, 6, (unsigned)CS, 1.f/(6.f*S_));
    }
}